// deformation_field_89077621719628
// MI455X (gfx1250) — compile-verified
//
#include <hip/hip_runtime.h>

#define N_THREADS 256
#define PAIRS 4                  // point-pairs per thread => 8 points per thread
#define GRID_W 64                // control grid is 64x64 (N_CTRL=62 + 2)
#define NPIX (GRID_W * GRID_W)   // 4096 pixels per channel

// reflect-101 (period 126) then clamp. For this kernel the raw index is always
// in [-2, 64] (|coord| <= 1 -> ix in [-0.5, 62.6]), so abs + one fold is exactly
// equivalent to the reference's abs % 126 -> fold -> clip over the reachable range.
__device__ __forceinline__ int reflect64(int i) {
    i = abs(i);
    i = (i >= GRID_W) ? (126 - i) : i;
    return min(max(i, 0), GRID_W - 1);
}

// Keys cubic weights, A = -0.75, same expression order as the reference
__device__ __forceinline__ void cubic_w(float t, float w[4]) {
    const float Ac = -0.75f;
    float s = 1.0f - t;
    w[0] = Ac * t * s * s;
    w[1] = (Ac + 2.0f) * t * t * t - (Ac + 3.0f) * t * t + 1.0f;
    w[2] = (Ac + 2.0f) * s * s * s - (Ac + 3.0f) * s * s + 1.0f;
    w[3] = Ac * s * t * t;
}

// One bicubic sample of the channel-interleaved LDS image.
// c0 = coordinate[:,0] (-> gy), c1 = coordinate[:,1] (-> gx)
__device__ __forceinline__ void sample_point(const float* __restrict__ s_img,
                                             float c0, float c1,
                                             float& o0, float& o1) {
    const float CROP = 1.0f / 64.0f;
    float gy = c0 - CROP;
    float gx = c1 - CROP;
    float ixf = (gx + 1.0f) * 0.5f * 63.0f;
    float iyf = (gy + 1.0f) * 0.5f * 63.0f;
    float fx = floorf(ixf), fy = floorf(iyf);
    float tx = ixf - fx,    ty = iyf - fy;
    int ix0 = (int)fx, iy0 = (int)fy;

    int xs[4], ys[4];
#pragma unroll
    for (int j = 0; j < 4; ++j) {
        xs[j] = reflect64(ix0 - 1 + j);
        ys[j] = reflect64(iy0 - 1 + j);
    }
    float wx[4], wy[4];
    cubic_w(tx, wx);
    cubic_w(ty, wy);

    float a0 = 0.0f, a1 = 0.0f;
#pragma unroll
    for (int i = 0; i < 4; ++i) {
        const float* row = s_img + (ys[i] << 7);   // y * 64 * 2
        float r0 = 0.0f, r1 = 0.0f;
#pragma unroll
        for (int j = 0; j < 4; ++j) {
            // one ds_load_b64 fetches both channels of tap (y, x)
            float2 v = *reinterpret_cast<const float2*>(row + (xs[j] << 1));
            r0 = fmaf(wx[j], v.x, r0);
            r1 = fmaf(wx[j], v.y, r1);
        }
        a0 = fmaf(wy[i], r0, a0);
        a1 = fmaf(wy[i], r1, a1);
    }
    o0 = a0;
    o1 = a1;
}

__global__ __launch_bounds__(N_THREADS)
void bicubic_defo_kernel(const float4* __restrict__ coord4,
                         const float* __restrict__ depl,
                         const float* __restrict__ mask,
                         float4* __restrict__ out4,
                         int npairs)
{
    // LDS image, channel-interleaved: s_img[(y*64 + x)*2 + c]  (32 KB)
    __shared__ float s_img[NPIX * 2];
    const int t = threadIdx.x;

    // ---- Stage control grid: async global->LDS, transposing (c,y,x)->(y,x,c).
    // Per-lane 4B scatter straight into the interleaved LDS slot, no VGPR
    // round-trip (CDNA5 GLOBAL_LOAD_ASYNC_TO_LDS, tracked by ASYNCcnt).
    {
        unsigned lds_base = (unsigned)(uintptr_t)(&s_img[0]); // low 32 bits = LDS offset
        for (int e = t; e < 2 * NPIX; e += N_THREADS) {
            int c = e >> 12;            // channel plane in global layout
            int p = e & (NPIX - 1);     // y*64 + x
            unsigned lds_addr = lds_base + (unsigned)((((unsigned)p << 1) | (unsigned)c) << 2);
            const float* g = depl + e;
            asm volatile("global_load_async_to_lds_b32 %0, %1, off"
                         :: "v"(lds_addr), "v"(g)
                         : "memory");
        }
        asm volatile("s_wait_asynccnt 0x0" ::: "memory");
    }
    __syncthreads();

    // ---- Apply mask in place (img = depl * mask, exactly as reference) ----
    for (int p = t; p < NPIX; p += N_THREADS) {
        float2 v = *reinterpret_cast<float2*>(&s_img[p << 1]);
        v.x *= mask[p];           // channel 0 plane
        v.y *= mask[NPIX + p];    // channel 1 plane
        *reinterpret_cast<float2*>(&s_img[p << 1]) = v;
    }
    __syncthreads();

    // ---- Per-point bicubic sampling: 2 points per float4, 4 float4s/thread ----
    const int base = blockIdx.x * (N_THREADS * PAIRS);

    float4 cbuf[PAIRS];
    bool ok[PAIRS];
#pragma unroll
    for (int k = 0; k < PAIRS; ++k) {
        int q = base + k * N_THREADS + t;
        ok[k] = (q < npairs);
        if (ok[k]) cbuf[k] = coord4[q];     // global_load_b128, batched up front
    }

#pragma unroll
    for (int k = 0; k < PAIRS; ++k) {
        if (ok[k]) {
            float4 co = cbuf[k];
            float4 o;
            sample_point(s_img, co.x, co.y, o.x, o.y);   // point 2q
            sample_point(s_img, co.z, co.w, o.z, o.w);   // point 2q+1
            out4[base + k * N_THREADS + t] = o;          // global_store_b128
        }
    }
}

extern "C" void kernel_launch(void* const* d_in, const int* in_sizes, int n_in,
                              void* d_out, int out_size, void* d_ws, size_t ws_size,
                              hipStream_t stream) {
    const float4* coord4 = (const float4*)d_in[0];  // (B, 2) float32 -> B/2 float4
    const float*  depl   = (const float*)d_in[1];   // (2, 64, 64) float32
    const float*  maskp  = (const float*)d_in[2];   // (2, 64, 64) float32
    float4* out4 = (float4*)d_out;                  // (B, 2) float32 -> B/2 float4

    int B = in_sizes[0] / 2;                        // B = 2097152 (even)
    int npairs = B / 2;
    int per_block = N_THREADS * PAIRS;
    int blocks = (npairs + per_block - 1) / per_block;

    bicubic_defo_kernel<<<blocks, N_THREADS, 0, stream>>>(coord4, depl, maskp, out4, npairs);
}